// Prompt_Generation_Model_9887014715496
// MI455X (gfx1250) — compile-verified
//
#include <hip/hip_runtime.h>
#include <hip/hip_bf16.h>
#include <math.h>

// Problem constants (from reference)
#define B_SZ      16
#define LK        8192
#define DMODEL    512      // N_HEAD(8) * D_K(64)
#define NHEAD     8
#define DK        64
#define FILTERNUM 10

typedef __attribute__((ext_vector_type(2))) float v2f;
typedef __attribute__((ext_vector_type(4))) float v4f;
typedef __attribute__((ext_vector_type(8))) float v8f;

// ---------------------------------------------------------------------------
// Kernel 1: scores[b, j] = sum_d q[b,d] * k[b,j,d]   via V_WMMA_F32_16X16X4_F32
//
// Per wave: 16-key tile. A = 16x4 f32 tile of k (ISA layout: lanes 0-15 supply
// K=0,1; lanes 16-31 supply K=2,3). B = q chunk replicated across all 16 N
// columns, which degenerates to the same per-lane 2-float pattern as A.
//
// K-chunk permutation trick (reduction over K is commutative across the two
// accumulating WMMAs): chunkA = dims {d,d+1,d+4,d+5}, chunkB = {d+2,d+3,d+6,d+7}.
// With per-lane base offset sh=(lane>>4)*4, ONE b128 load per lane per 8 dims
// supplies both fragments: elements (0,1) -> chunkA, (2,3) -> chunkB, for both
// half-waves, with zero lane-dependent selects. Same for the LDS q load.
//
// Extraction (C/D 16x16 f32 layout): lane 0 -> keys m=0..7 in c[0..7],
// lane 16 -> keys m=8..15 in c[0..7].
// ---------------------------------------------------------------------------
__global__ void score_kernel(const float* __restrict__ q,
                             const float* __restrict__ k,
                             float* __restrict__ scores) {
    const int b    = blockIdx.y;              // batch
    const int lane = threadIdx.x & 31;
    const int wave = threadIdx.x >> 5;        // 0..7 (256 threads = 8 waves)
    const int j0   = blockIdx.x * 128 + wave * 16;   // first key of this wave's tile

    __shared__ float qs[DMODEL];
    for (int i = threadIdx.x; i < DMODEL; i += blockDim.x)
        qs[i] = q[b * DMODEL + i];
    __syncthreads();

    const int row = lane & 15;                // key within tile (M)
    const int sh  = (lane >> 4) * 4;          // half-wave K base (floats)

    const float* kp = k + ((size_t)b * LK + (size_t)(j0 + row)) * DMODEL + sh;
    const float* qp = qs + sh;

    v8f c0 = {0.f,0.f,0.f,0.f,0.f,0.f,0.f,0.f};
    v8f c1 = {0.f,0.f,0.f,0.f,0.f,0.f,0.f,0.f};

    #pragma unroll 8
    for (int d = 0; d < DMODEL; d += 8) {
        v4f a  = __builtin_nontemporal_load((const v4f*)(kp + d));  // k: streamed once, NT
        v4f bq = *(const v4f*)(qp + d);                             // q from LDS (broadcast)

        v2f aA = __builtin_shufflevector(a,  a,  0, 1);
        v2f aB = __builtin_shufflevector(a,  a,  2, 3);
        v2f bA = __builtin_shufflevector(bq, bq, 0, 1);
        v2f bB = __builtin_shufflevector(bq, bq, 2, 3);

        c0 = __builtin_amdgcn_wmma_f32_16x16x4_f32(false, aA, false, bA,
                                                   (short)0, c0, false, false);
        c1 = __builtin_amdgcn_wmma_f32_16x16x4_f32(false, aB, false, bB,
                                                   (short)0, c1, false, false);
    }
    v8f c = c0 + c1;

    if ((lane & 15) == 0) {                   // lanes 0 and 16
        const int base = b * LK + j0 + (lane >> 4) * 8;
        #pragma unroll
        for (int r = 0; r < 8; ++r)
            scores[base + r] = c[r];
    }
}

// ---------------------------------------------------------------------------
// Kernel 2: per-batch top-10 indices via 10 rounds of block-wide argmax.
// Each of 256 threads caches 32 contiguous scores in registers; the chosen
// element is poisoned locally each round. (Final output is invariant to the
// order of the selected set.)
// ---------------------------------------------------------------------------
__global__ void topk_kernel(const float* __restrict__ scores,
                            int* __restrict__ topidx) {
    const int b = blockIdx.x;
    const int t = threadIdx.x;            // 256 threads

    float local[32];
    const float* s = scores + (size_t)b * LK + t * 32;
    #pragma unroll
    for (int i = 0; i < 32; ++i) local[i] = s[i];

    __shared__ float sval[256];
    __shared__ int   sidx[256];
    __shared__ int   chosen;

    for (int round = 0; round < FILTERNUM; ++round) {
        float best = -INFINITY; int bi = 0;
        #pragma unroll
        for (int i = 0; i < 32; ++i)
            if (local[i] > best) { best = local[i]; bi = i; }
        sval[t] = best;
        sidx[t] = t * 32 + bi;
        __syncthreads();

        for (int off = 128; off > 0; off >>= 1) {
            if (t < off) {
                const float vo = sval[t + off];
                const int   io = sidx[t + off];
                if (vo > sval[t] || (vo == sval[t] && io < sidx[t])) {
                    sval[t] = vo; sidx[t] = io;
                }
            }
            __syncthreads();
        }
        if (t == 0) {
            chosen = sidx[0];
            topidx[b * FILTERNUM + round] = chosen;
        }
        __syncthreads();
        const int c = chosen;
        if ((c >> 5) == t) local[c & 31] = -INFINITY;
        __syncthreads();
    }
}

// ---------------------------------------------------------------------------
// Kernel 3: gather 10 k/v rows, 8-head softmax attention over 10 keys.
// One 512-thread block per batch; tiny, entirely LDS-resident after gather.
// ---------------------------------------------------------------------------
__global__ void attn_kernel(const float* __restrict__ q,
                            const float* __restrict__ k,
                            const float* __restrict__ v,
                            const int* __restrict__ topidx,
                            float* __restrict__ out) {
    const int b = blockIdx.x;
    const int t = threadIdx.x;            // 512 threads == DMODEL

    __shared__ float fk[FILTERNUM][DMODEL];
    __shared__ float fv[FILTERNUM][DMODEL];
    __shared__ float sc[NHEAD][FILTERNUM];
    __shared__ float pr[NHEAD][FILTERNUM];

    #pragma unroll
    for (int j = 0; j < FILTERNUM; ++j) {
        const int idx = topidx[b * FILTERNUM + j];
        const size_t roff = ((size_t)b * LK + (size_t)idx) * DMODEL + t;
        fk[j][t] = k[roff];
        fv[j][t] = v[roff];
    }
    __syncthreads();

    if (t < NHEAD * FILTERNUM) {          // 80 threads: logits
        const int h = t / FILTERNUM, j = t % FILTERNUM;
        float s = 0.f;
        #pragma unroll
        for (int d = 0; d < DK; ++d)
            s += q[b * DMODEL + h * DK + d] * fk[j][h * DK + d];
        sc[h][j] = s;
    }
    __syncthreads();

    if (t < NHEAD) {                      // 8 threads: softmax over 10
        float m = sc[t][0];
        #pragma unroll
        for (int j = 1; j < FILTERNUM; ++j) m = fmaxf(m, sc[t][j]);
        float sum = 0.f;
        #pragma unroll
        for (int j = 0; j < FILTERNUM; ++j) {
            const float e = __expf(sc[t][j] - m);
            pr[t][j] = e; sum += e;
        }
        const float inv = 1.f / sum;
        #pragma unroll
        for (int j = 0; j < FILTERNUM; ++j) pr[t][j] *= inv;
    }
    __syncthreads();

    const int h = t >> 6;                 // head of this output dim
    float acc = 0.f;
    #pragma unroll
    for (int j = 0; j < FILTERNUM; ++j)
        acc += pr[h][j] * fv[j][t];
    out[b * DMODEL + t] = acc;
}

// ---------------------------------------------------------------------------
extern "C" void kernel_launch(void* const* d_in, const int* in_sizes, int n_in,
                              void* d_out, int out_size, void* d_ws, size_t ws_size,
                              hipStream_t stream) {
    const float* q = (const float*)d_in[0];   // [16, 1, 512]
    const float* k = (const float*)d_in[1];   // [16, 8192, 512]
    const float* v = (const float*)d_in[2];   // [16, 8192, 512]
    float* out = (float*)d_out;               // [16, 512]

    float* scores = (float*)d_ws;                                  // 16*8192 f32 = 512 KB
    int*   topidx = (int*)((char*)d_ws + (size_t)B_SZ * LK * sizeof(float)); // 160 ints

    dim3 g1(LK / 128, B_SZ);                  // 64 x 16 blocks, 8 waves each
    score_kernel<<<g1, 256, 0, stream>>>(q, k, scores);
    topk_kernel<<<B_SZ, 256, 0, stream>>>(scores, topidx);
    attn_kernel<<<B_SZ, DMODEL, 0, stream>>>(q, k, v, topidx, out);
}